// Kwinners_41214506173086
// MI455X (gfx1250) — compile-verified
//
#include <hip/hip_runtime.h>
#include <hip/hip_bf16.h>

#define NCOLS   4096
#define KSEL    512u
#define THREADS 256
#define WAVES   8          // wave32: 256/32
#define HSTRIDE 257        // 257 % 64 == 1 -> per-wave copies of a bin hit different LDS banks

#if defined(__HIP_DEVICE_COMPILE__) && __has_builtin(__builtin_amdgcn_global_load_async_to_lds_b128)
#define USE_ASYNC_LDS 1
#else
#define USE_ASYNC_LDS 0
#endif

typedef int v4i __attribute__((vector_size(16)));
typedef __attribute__((address_space(1))) v4i* gv4i_ptr;   // global (prints as __device__)
typedef __attribute__((address_space(3))) v4i* lv4i_ptr;   // LDS    (prints as __shared__)

// Monotonic bijection f32 -> u32 (larger float => larger key), and exact inverse.
__device__ __forceinline__ unsigned f2key(float f) {
    unsigned u = __float_as_uint(f);
    return u ^ (unsigned)(((int)u >> 31) | 0x80000000);
}
__device__ __forceinline__ float key2f(unsigned k) {
    unsigned u = (k & 0x80000000u) ? (k ^ 0x80000000u) : ~k;
    return __uint_as_float(u);
}

__global__ __launch_bounds__(THREADS) void kwinners_radix_select(
    const float* __restrict__ in, float* __restrict__ out) {

#if USE_ASYNC_LDS
    __shared__ float4 sVals[NCOLS / 4];        // 16 KB row staging (filled by async DMA path)
#endif
    __shared__ unsigned sHist[WAVES * HSTRIDE]; // ~8 KB per-wave privatized histograms
    __shared__ unsigned sSel[2];                // {selected digit, count strictly above}

    const int tid  = threadIdx.x;
    const int wave = tid >> 5;                  // wave32
    const size_t rowBase = (size_t)blockIdx.x * NCOLS;
    const float4* __restrict__ in4  = (const float4*)(in  + rowBase);
    float4* __restrict__       out4 = (float4*)(out + rowBase);

#if USE_ASYNC_LDS
    // gfx1250 async global->LDS: each lane DMAs one 16B chunk per issue; 4 issues cover the row.
    #pragma unroll
    for (int s = 0; s < 4; ++s) {
        const int vi = tid + THREADS * s;
        __builtin_amdgcn_global_load_async_to_lds_b128(
            (gv4i_ptr)&in4[vi],
            (lv4i_ptr)&sVals[vi],
            0, 0);
    }
#if __has_builtin(__builtin_amdgcn_s_wait_asynccnt)
    __builtin_amdgcn_s_wait_asynccnt(0);
#else
    asm volatile("s_wait_asynccnt 0" ::: "memory");
#endif
    __syncthreads();
#endif

    // Pull this thread's 16 elements and convert to sortable keys (register-resident).
    unsigned key[16];
    #pragma unroll
    for (int s = 0; s < 4; ++s) {
        const int vi = tid + THREADS * s;
#if USE_ASYNC_LDS
        float4 v = sVals[vi];
#else
        float4 v = in4[vi];
#endif
        key[4*s+0] = f2key(v.x);
        key[4*s+1] = f2key(v.y);
        key[4*s+2] = f2key(v.z);
        key[4*s+3] = f2key(v.w);
    }

    // Exact MSB-first radix select of the K-th largest key (4 passes x 8 bits).
    unsigned prefix = 0, pmask = 0, k = KSEL;

    #pragma unroll
    for (int shift = 24; shift >= 0; shift -= 8) {
        for (int i = tid; i < WAVES * HSTRIDE; i += THREADS) sHist[i] = 0;
        __syncthreads();

        unsigned* myh = &sHist[wave * HSTRIDE];
        #pragma unroll
        for (int s = 0; s < 16; ++s) {
            if ((key[s] & pmask) == prefix)
                atomicAdd(&myh[(key[s] >> shift) & 255u], 1u);
        }
        __syncthreads();

        if (tid < 32) {  // wave 0: reduce 8 histograms + suffix-scan 256 bins, no barriers
            unsigned h[8], T = 0;
            #pragma unroll
            for (int i = 0; i < 8; ++i) {
                const unsigned b = (unsigned)tid * 8u + (unsigned)i;
                unsigned v = 0;
                #pragma unroll
                for (int w = 0; w < WAVES; ++w) v += sHist[w * HSTRIDE + b];
                h[i] = v; T += v;
            }
            // inclusive suffix scan of per-lane chunk totals across the wave
            unsigned S = T;
            #pragma unroll
            for (int off = 1; off < 32; off <<= 1) {
                unsigned o = __shfl_down(S, off, 32);
                if (tid + off < 32) S += o;
            }
            unsigned cum = S - T;  // candidates in bins strictly above this lane's chunk
            #pragma unroll
            for (int i = 7; i >= 0; --i) {
                const unsigned b = (unsigned)tid * 8u + (unsigned)i;
                const unsigned c = h[i];
                if (cum < k && k <= cum + c) { sSel[0] = b; sSel[1] = cum; }
                cum += c;
            }
        }
        __syncthreads();

        prefix |= sSel[0] << shift;
        pmask  |= 0xFFu << shift;
        k      -= sSel[1];
        // (two barriers separate these sSel reads from the next pass's sSel write)
    }

    // prefix == exact key of the K-th largest element; keep key >= prefix (matches s >= thresh).
    const unsigned th = prefix;
    #pragma unroll
    for (int s = 0; s < 4; ++s) {
        const int vi = tid + THREADS * s;
        float4 o;
        o.x = (key[4*s+0] >= th) ? key2f(key[4*s+0]) : 0.0f;
        o.y = (key[4*s+1] >= th) ? key2f(key[4*s+1]) : 0.0f;
        o.z = (key[4*s+2] >= th) ? key2f(key[4*s+2]) : 0.0f;
        o.w = (key[4*s+3] >= th) ? key2f(key[4*s+3]) : 0.0f;
        out4[vi] = o;
    }
}

extern "C" void kernel_launch(void* const* d_in, const int* in_sizes, int n_in,
                              void* d_out, int out_size, void* d_ws, size_t ws_size,
                              hipStream_t stream) {
    (void)n_in; (void)d_ws; (void)ws_size; (void)out_size;
    const float* s = (const float*)d_in[0];
    float* o = (float*)d_out;
    const int rows = in_sizes[0] / NCOLS;   // 16384
    kwinners_radix_select<<<rows, THREADS, 0, stream>>>(s, o);
}